// D_SUM_CALC_25202868092936
// MI455X (gfx1250) — compile-verified
//
#include <hip/hip_runtime.h>

// ---------------------------------------------------------------------------
// D_SUM via 2D summed-area table (SAT), CDNA5 / gfx1250, wave32.
//
// out[b,i,j] = block-sum of D over [min(i,j)..max(i,j)]^2, except the first
// subdiagonal which is raw D.  SAT P[r,s] = sum_{p<=r,q<=s} D gives each
// output as 4 gathers (inclusion-exclusion).
//
// Kernel 1 does the row-wise inclusive scan with V_WMMA_F32_16X16X4_F32:
// for each 16x16 tile T,   Tprefix = T * U16 + carry_broadcast
// where U16 is upper-triangular ones (exact in f32 WMMA), carry is the
// running row total fed in through the C accumulator.
// ---------------------------------------------------------------------------

typedef __attribute__((ext_vector_type(2))) float v2f;
typedef __attribute__((ext_vector_type(8))) float v8f;

#define N 2048          // fixed by the reference (n = 2048)
#define NTILE (N / 16)  // 128 tiles along a row
#define LOG2N 11

// ---------------------------------------------------------------------------
// Kernel 1: row-wise inclusive prefix sums. One wave32 per 16-row strip.
// Strips: B * (N/16).  8 waves per 256-thread block.
// ---------------------------------------------------------------------------
__global__ __launch_bounds__(256) void rowscan_wmma(const float* __restrict__ D,
                                                    float* __restrict__ R) {
  const int lane = threadIdx.x & 31;
  const int wave = threadIdx.x >> 5;
  const int strip = blockIdx.x * 8 + wave;   // 0 .. B*(N/16)-1
  const int b     = strip >> 7;              // 128 strips per batch
  const int row0  = (strip & 127) << 4;

  const float* __restrict__ Db = D + (size_t)b * N * N;
  float* __restrict__       Rb = R + (size_t)b * N * N;

  __shared__ float lds[8][16 * 16];          // 1 KiB per wave
  float* tl = lds[wave];

  const int m    = lane & 15;                // A-row / B-col / D-col for this lane
  const int half = lane >> 4;                // lane half selects K pair / M+8

  // C accumulator doubles as the broadcast row-carry (starts at 0).
  v8f c = {0.f, 0.f, 0.f, 0.f, 0.f, 0.f, 0.f, 0.f};

  for (int tile = 0; tile < NTILE; ++tile) {
    // ---- stage 16x16 tile into LDS, coalesced 128-bit loads --------------
#pragma unroll
    for (int t = 0; t < 2; ++t) {
      const int idx4 = t * 32 + lane;        // 0..63 float4s
      const int r    = idx4 >> 2;
      const int cg   = idx4 & 3;
      const float4 v =
          *(const float4*)(Db + (size_t)(row0 + r) * N + tile * 16 + cg * 4);
      *(float4*)&tl[r * 16 + cg * 4] = v;
    }
    // same-wave LDS RAW: compiler inserts s_wait_dscnt; no barrier needed
    // (each wave owns its private LDS region).

    // ---- Tprefix = T * U16 + carry : 4 chained K=4 f32 WMMAs -------------
    // A 16x4 f32 layout: lanes 0-15 -> {K=k0, K=k0+1}, lanes 16-31 -> {k0+2, k0+3}.
    // B 4x16 f32 layout mirrors A's K pairing; B col = lane&15.
#pragma unroll
    for (int s = 0; s < 4; ++s) {
      const int k0 = s * 4 + half * 2;
      v2f a, u;
      a.x = tl[m * 16 + k0];
      a.y = tl[m * 16 + k0 + 1];
      u.x = (k0     <= m) ? 1.0f : 0.0f;     // U16[q][j] = (q <= j), j = lane&15
      u.y = (k0 + 1 <= m) ? 1.0f : 0.0f;
      // 8 args: (neg_a, A, neg_b, B, c_mod, C, reuse_a, reuse_b)
      c = __builtin_amdgcn_wmma_f32_16x16x4_f32(
          /*neg_a=*/false, a, /*neg_b=*/false, u,
          /*c_mod=*/(short)0, c, /*reuse_a=*/false, /*reuse_b=*/false);
    }

    // ---- store row-prefix tile (D layout: vgpr r -> row r + 8*half) ------
    const int col = tile * 16 + m;
#pragma unroll
    for (int r = 0; r < 8; ++r) {
      const int row = row0 + r + half * 8;
      Rb[(size_t)row * N + col] = c[r];
    }

    // ---- new carry = inclusive prefix at column 15, broadcast per half ---
    const int src = half ? 31 : 15;
#pragma unroll
    for (int r = 0; r < 8; ++r) c[r] = __shfl(c[r], src, 32);
  }
}

// ---------------------------------------------------------------------------
// Kernel 2: in-place column-wise accumulation -> full SAT.
// One thread per (batch, column); fully coalesced walk down the rows.
// ---------------------------------------------------------------------------
__global__ __launch_bounds__(256) void colscan(float* __restrict__ P) {
  const int colg = blockIdx.x * blockDim.x + threadIdx.x;  // 0 .. B*N-1
  const int b = colg >> LOG2N;
  const int j = colg & (N - 1);
  float* base = P + (size_t)b * N * N + j;

  float acc = 0.0f;
#pragma unroll 4
  for (int i = 0; i < N; ++i) {
    __builtin_prefetch(base + (size_t)(i + 16) * N, 0, 0);  // global_prefetch
    const float v = base[(size_t)i * N];
    acc += v;
    base[(size_t)i * N] = acc;
  }
}

// ---------------------------------------------------------------------------
// Kernel 3: assemble output from the SAT (4-point inclusion-exclusion),
// with the raw-D subdiagonal override.  P gathers hit the 192 MB L2.
// ---------------------------------------------------------------------------
__global__ __launch_bounds__(256) void assemble(const float* __restrict__ D,
                                                const float* __restrict__ P,
                                                float* __restrict__ out) {
  const size_t t = (size_t)blockIdx.x * blockDim.x + threadIdx.x;
  const int j = (int)(t & (N - 1));
  const int i = (int)((t >> LOG2N) & (N - 1));
  const int b = (int)(t >> (2 * LOG2N));
  const float* __restrict__ Pb = P + (size_t)b * N * N;

  float r;
  if (i == j + 1) {
    // first subdiagonal keeps the raw input value
    r = D[(size_t)b * N * N + (size_t)i * N + j];
  } else {
    const int a = (i < j) ? i : j;
    const int cmx = (i < j) ? j : i;
    r = Pb[(size_t)cmx * N + cmx];
    if (a > 0) {
      r -= Pb[(size_t)(a - 1) * N + cmx];
      r -= Pb[(size_t)cmx * N + (a - 1)];
      r += Pb[(size_t)(a - 1) * N + (a - 1)];
    }
  }
  out[t] = r;
}

// ---------------------------------------------------------------------------
extern "C" void kernel_launch(void* const* d_in, const int* in_sizes, int n_in,
                              void* d_out, int out_size, void* d_ws, size_t ws_size,
                              hipStream_t stream) {
  const float* D = (const float*)d_in[0];
  float* out = (float*)d_out;
  float* P   = (float*)d_ws;                       // needs B*N*N*4 = 128 MiB
  const int B = in_sizes[0] / (N * N);             // 8

  // 1) row-wise inclusive scan (WMMA): D -> P(row prefixes)
  rowscan_wmma<<<dim3((B * NTILE) / 8), dim3(256), 0, stream>>>(D, P);
  // 2) column-wise accumulate in place: P -> full SAT
  colscan<<<dim3((B * N) / 256), dim3(256), 0, stream>>>(P);
  // 3) gather SAT into the output
  const size_t total = (size_t)B * N * N;
  assemble<<<dim3((unsigned)(total / 256)), dim3(256), 0, stream>>>(D, P, out);
}